// Model_PPI_16406775071386
// MI455X (gfx1250) — compile-verified
//
#include <hip/hip_runtime.h>
#include <math.h>

typedef __attribute__((ext_vector_type(2))) float v2f;
typedef __attribute__((ext_vector_type(8))) float v8f;

#define NN 50000
#define NE 800000
#define ET (NE + NN)          // edges incl. self-loops
#define NEG_SLOPE 0.2f

// ---------------------------------------------------------------------------
// GEMM: XW[M,Nout] = X[M,K] * W[K,Nout], row-major, fp32 WMMA 16x16x4.
// grid.x = M/16 (M % 16 == 0). NT = #16-wide column tiles, block = NT waves.
// A tile staged via CDNA5 async global->LDS copies (ASYNCcnt tracked).
// Out-of-range columns are CLAMPED (not zeroed): their WMMA results only feed
// their own (never-stored) output columns, so no predication is needed and
// EXEC stays all-ones as the WMMA ISA requires.
// ---------------------------------------------------------------------------
template<int NT, int K>
__global__ __launch_bounds__(NT * 32)
void gemm_wmma_f32(const float* __restrict__ X, const float* __restrict__ W,
                   float* __restrict__ XW, int Nout) {
    constexpr int KC  = 32;              // K chunk
    constexpr int LDA = 36;              // row stride: 16B-aligned, conflict-free
    __shared__ float As[16 * LDA];

    const int m0   = blockIdx.x * 16;
    const int tid  = threadIdx.x;
    const int lane = tid & 31;
    const int half = lane >> 4;          // 0: lanes 0-15, 1: lanes 16-31
    const int lrow = lane & 15;
    const int col  = (tid >> 5) * 16 + lrow;
    const bool colOK = col < Nout;
    const int  ccol  = colOK ? col : Nout - 1;     // clamped, branch-free loads

    // tile movers: first 128 threads, one b128 (float4) each: 16 rows x 32 cols
    const int  lr     = tid >> 3;        // row 0..15
    const int  lc     = (tid & 7) * 4;   // col 0,4,...,28
    const bool loader = tid < 128;
    const unsigned ldsDst = (unsigned)(size_t)&As[lr * LDA + lc];

    v8f acc = {};
    for (int k0 = 0; k0 < K; k0 += KC) {
        if (loader) {
            unsigned goff = (unsigned)(((m0 + lr) * K + k0 + lc) * sizeof(float));
            asm volatile("global_load_async_to_lds_b128 %0, %1, %2 offset:0"
                         :: "v"(ldsDst), "v"(goff), "s"(X) : "memory");
        }
        asm volatile("s_wait_asynccnt 0" ::: "memory");
        __syncthreads();
        #pragma unroll
        for (int kk = 0; kk < KC; kk += 4) {
            // A 16x4 frag: lanes 0-15 hold K=0,1 ; lanes 16-31 hold K=2,3
            const float2 af = *(const float2*)&As[lrow * LDA + kk + 2 * half];
            v2f a, b;
            a.x = af.x; a.y = af.y;
            // B 4x16 frag: VGPR v <-> K = 2*half + v, N = lane%16
            const int krow = k0 + kk + 2 * half;
            b.x = W[(size_t)(krow + 0) * Nout + ccol];
            b.y = W[(size_t)(krow + 1) * Nout + ccol];
            acc = __builtin_amdgcn_wmma_f32_16x16x4_f32(
                false, a, false, b, (short)0, acc, false, false);
        }
        __syncthreads();
    }
    if (colOK) {
        #pragma unroll
        for (int r = 0; r < 8; ++r)                    // C/D: M = r + 8*half
            XW[(size_t)(m0 + r + 8 * half) * Nout + col] = acc[r];
    }
}

// ---------------------------------------------------------------------------
// per-node attention projections: alpha_s/d[n,h] = sum_c xw[n,h,c]*a[h,c]
// ---------------------------------------------------------------------------
__global__ void alpha_kernel(const float* __restrict__ xw,
                             const float* __restrict__ a_src,
                             const float* __restrict__ a_dst,
                             float* __restrict__ as, float* __restrict__ ad,
                             int H, int C) {
    int idx = blockIdx.x * blockDim.x + threadIdx.x;
    if (idx >= NN * H) return;
    int n = idx / H, h = idx - n * H;
    const float* row = xw + (size_t)n * H * C + h * C;
    float s = 0.f, d = 0.f;
    for (int c = 0; c < C; ++c) {
        s += row[c] * a_src[h * C + c];
        d += row[c] * a_dst[h * C + c];
    }
    as[idx] = s; ad[idx] = d;
}

__global__ void fill_kernel(float* __restrict__ p, float v, int n) {
    int i = blockIdx.x * blockDim.x + threadIdx.x;
    if (i < n) p[i] = v;
}

__device__ __forceinline__ void edge_sd(const int* __restrict__ ei, int e,
                                        int& s, int& d) {
    if (e < NE) { s = ei[e]; d = ei[NE + e]; }
    else        { s = d = e - NE; }              // appended self-loops
}

__device__ __forceinline__ float leaky(float v) {
    return v > 0.f ? v : NEG_SLOPE * v;
}

// monotonic float atomic-max: signed-int max for >=0, unsigned min for <0
__device__ __forceinline__ void atomicMaxFloat(float* addr, float v) {
    int iv = __float_as_int(v);
    if (iv >= 0) atomicMax((int*)addr, iv);
    else         atomicMin((unsigned int*)addr, (unsigned int)iv);
}

__global__ void edge_max(const int* __restrict__ ei, const float* __restrict__ as,
                         const float* __restrict__ ad, float* __restrict__ emax,
                         int H) {
    int idx = blockIdx.x * blockDim.x + threadIdx.x;
    if (idx >= ET * H) return;
    int e = idx / H, h = idx - e * H;
    int s, d; edge_sd(ei, e, s, d);
    atomicMaxFloat(&emax[d * H + h], leaky(as[s * H + h] + ad[d * H + h]));
}

__global__ void edge_sum(const int* __restrict__ ei, const float* __restrict__ as,
                         const float* __restrict__ ad, const float* __restrict__ emax,
                         float* __restrict__ denom, int H) {
    int idx = blockIdx.x * blockDim.x + threadIdx.x;
    if (idx >= ET * H) return;
    int e = idx / H, h = idx - e * H;
    int s, d; edge_sd(ei, e, s, d);
    float v = leaky(as[s * H + h] + ad[d * H + h]);
    atomicAdd(&denom[d * H + h], __expf(v - emax[d * H + h]));
}

// weighted message scatter; each thread handles CV consecutive channels
__global__ void edge_msg(const int* __restrict__ ei, const float* __restrict__ as,
                         const float* __restrict__ ad, const float* __restrict__ emax,
                         const float* __restrict__ denom,
                         const float* __restrict__ xw, float* __restrict__ out,
                         int H, int C, int CV) {
    int nc = (C + CV - 1) / CV;
    int total = ET * H * nc;
    int idx = blockIdx.x * blockDim.x + threadIdx.x;
    if (idx >= total) return;
    int t = idx;
    int cc = t % nc; t /= nc;
    int h  = t % H;  int e = t / H;
    int s, d; edge_sd(ei, e, s, d);
    float v = leaky(as[s * H + h] + ad[d * H + h]);
    float alpha = __expf(v - emax[d * H + h]) / (denom[d * H + h] + 1e-16f);
    int c0 = cc * CV, c1 = c0 + CV; if (c1 > C) c1 = C;
    const float* xs = xw  + (size_t)s * H * C + h * C;
    float*       od = out + (size_t)d * H * C + h * C;
    for (int c = c0; c < c1; ++c) atomicAdd(&od[c], xs[c] * alpha);
}

__global__ void bias_relu(const float* __restrict__ acc, const float* __restrict__ b,
                          float* __restrict__ out, int F) {
    int idx = blockIdx.x * blockDim.x + threadIdx.x;
    if (idx >= NN * F) return;
    float v = acc[idx] + b[idx % F];
    out[idx] = v > 0.f ? v : 0.f;
}

// layer-3 finalize: relu(acc+b) then softmax over class pairs -> d_out
__global__ void final_softmax(const float* __restrict__ acc,
                              const float* __restrict__ b,
                              float* __restrict__ out) {
    int idx = blockIdx.x * blockDim.x + threadIdx.x;
    if (idx >= NN * 121) return;
    int n = idx / 121, k = idx - n * 121;
    size_t base = (size_t)n * 242 + 2 * k;
    float t0 = acc[base + 0] + b[2 * k + 0]; t0 = t0 > 0.f ? t0 : 0.f;
    float t1 = acc[base + 1] + b[2 * k + 1]; t1 = t1 > 0.f ? t1 : 0.f;
    float m  = fmaxf(t0, t1);
    float e0 = __expf(t0 - m), e1 = __expf(t1 - m);
    float inv = 1.f / (e0 + e1);
    out[base + 0] = e0 * inv;
    out[base + 1] = e1 * inv;
}

// ---------------------------------------------------------------------------
static inline int g256(int n) { return (n + 255) / 256; }

static void gat_edges(const float* avsW, const float* avdW, int H, int C,
                      const int* ei, const float* XW, float* OT,
                      float* AS, float* AD, float* EM, float* DN,
                      hipStream_t stream) {
    alpha_kernel<<<g256(NN * H), 256, 0, stream>>>(XW, avsW, avdW, AS, AD, H, C);
    fill_kernel<<<g256(NN * H), 256, 0, stream>>>(EM, -INFINITY, NN * H);
    fill_kernel<<<g256(NN * H), 256, 0, stream>>>(DN, 0.f, NN * H);
    fill_kernel<<<g256(NN * H * C), 256, 0, stream>>>(OT, 0.f, NN * H * C);
    edge_max<<<g256(ET * H), 256, 0, stream>>>(ei, AS, AD, EM, H);
    edge_sum<<<g256(ET * H), 256, 0, stream>>>(ei, AS, AD, EM, DN, H);
    const int CV = 8;
    int nc = (C + CV - 1) / CV;
    edge_msg<<<g256(ET * H * nc), 256, 0, stream>>>(ei, AS, AD, EM, DN, XW, OT, H, C, CV);
}

extern "C" void kernel_launch(void* const* d_in, const int* in_sizes, int n_in,
                              void* d_out, int out_size, void* d_ws, size_t ws_size,
                              hipStream_t stream) {
    const float* x   = (const float*)d_in[0];
    const int*   ei  = (const int*)d_in[1];
    const float* W1  = (const float*)d_in[2];
    const float* a1s = (const float*)d_in[3];
    const float* a1d = (const float*)d_in[4];
    const float* b1  = (const float*)d_in[5];
    const float* W2  = (const float*)d_in[6];
    const float* a2s = (const float*)d_in[7];
    const float* a2d = (const float*)d_in[8];
    const float* b2  = (const float*)d_in[9];
    const float* W3  = (const float*)d_in[10];
    const float* a3s = (const float*)d_in[11];
    const float* a3d = (const float*)d_in[12];
    const float* b3  = (const float*)d_in[13];
    float* out = (float*)d_out;

    float* ws = (float*)d_ws;
    float* hA = ws;                              // [NN,64]
    float* hB = hA + (size_t)NN * 64;            // [NN,64]
    float* XW = hB + (size_t)NN * 64;            // [NN,242] max
    float* OT = XW + (size_t)NN * 242;           // [NN,242] max
    float* AS = OT + (size_t)NN * 242;           // [NN,8]
    float* AD = AS + (size_t)NN * 8;
    float* EM = AD + (size_t)NN * 8;
    float* DN = EM + (size_t)NN * 8;

    // Layer 1: 128 -> 8 heads x 8
    gemm_wmma_f32<4, 128><<<NN / 16, 4 * 32, 0, stream>>>(x, W1, XW, 64);
    gat_edges(a1s, a1d, 8, 8, ei, XW, OT, AS, AD, EM, DN, stream);
    bias_relu<<<g256(NN * 64), 256, 0, stream>>>(OT, b1, hA, 64);

    // Layer 2: 64 -> 8 heads x 8
    gemm_wmma_f32<4, 64><<<NN / 16, 4 * 32, 0, stream>>>(hA, W2, XW, 64);
    gat_edges(a2s, a2d, 8, 8, ei, XW, OT, AS, AD, EM, DN, stream);
    bias_relu<<<g256(NN * 64), 256, 0, stream>>>(OT, b2, hB, 64);

    // Layer 3: 64 -> 1 head x 242, mean over 1 head == identity, then softmax
    gemm_wmma_f32<16, 64><<<NN / 16, 16 * 32, 0, stream>>>(hB, W3, XW, 242);
    gat_edges(a3s, a3d, 1, 242, ei, XW, OT, AS, AD, EM, DN, stream);
    final_softmax<<<g256(NN * 121), 256, 0, stream>>>(OT, b3, out);
}